// GraphAttentionLayer_63548336112121
// MI455X (gfx1250) — compile-verified
//
#include <hip/hip_runtime.h>
#include <hip/hip_bf16.h>

// ---------------------------------------------------------------------------
// GAT layer for MI455X (gfx1250, wave32).
//   N=4096, D_IN=512, D_OUT=64, H=8 heads, F=8 head_dim, alpha=0.2
//
// Pipeline:
//   1) gat_linear : h = x @ W^T + b          (WMMA f32 16x16x4, TDM x-tile load)
//   2) gat_e      : e0/e1 per (node, head)   (tiny VALU)
//   3) gat_pack   : adj (64MB int32) -> 2MB bitmask (L2-resident)
//   4) gat_attn   : flash-style online-softmax attention + aggregation,
//                   one wave per head, WMMA f32 16x16x4 for P @ h_j,
//                   ballot-uniform max-skip for the rescale path
// ---------------------------------------------------------------------------

typedef __attribute__((ext_vector_type(2))) float        v2f;
typedef __attribute__((ext_vector_type(8))) float        v8f;
typedef __attribute__((ext_vector_type(4))) unsigned int v4u;
typedef __attribute__((ext_vector_type(4))) int          v4i;
typedef __attribute__((ext_vector_type(8))) int          v8i;

#define GAT_N     4096
#define GAT_DIN   512
#define GAT_DOUT  64
#define GAT_H     8
#define GAT_F     8
#define GAT_ALPHA 0.2f
#define XS_STRIDE 516   // 512 + 2-dword pad every 256 dwords (TDM pad feature)

// LDS index for the x tile: pad 2 dwords after every 256 dwords.
// Matches TDM pad_interval=256 dwords / pad_amount=2 dwords exactly, and keeps
// the 16-lane A-operand column reads conflict-free across 64 LDS banks.
__device__ __forceinline__ int xs_idx(int row, int col) {
    return row * XS_STRIDE + col + ((col >> 8) << 1);
}

// ---------------------------------------------------------------------------
// Kernel 1: h = x @ W^T + b.  Grid: 256 blocks (16 rows each), 128 thr (4 waves).
// Wave w computes the 16x16 tile of columns [16w, 16w+16).
// x tile [16 x 512] staged in LDS by the Tensor Data Mover (fallback: manual).
// ---------------------------------------------------------------------------
__global__ __launch_bounds__(128) void gat_linear(const float* __restrict__ x,
                                                  const float* __restrict__ W,
                                                  const float* __restrict__ bias,
                                                  float* __restrict__ hout) {
    __shared__ float xs[16 * XS_STRIDE];
    const int ib  = blockIdx.x * 16;
    const int tid = threadIdx.x;

#if defined(__has_builtin) && __has_builtin(__builtin_amdgcn_tensor_load_to_lds)
    // --- TDM path: one wave issues a 2D tile DMA (16 rows x 512 f32) --------
    if (tid < 32) {
        const unsigned long long gaddr =
            (unsigned long long)(const void*)(x + (size_t)ib * GAT_DIN);
        const unsigned ldsa = (unsigned)(unsigned long long)(void*)&xs[0];

        v4u g0;
        g0[0] = 1u;                                  // count=1, user descriptor
        g0[1] = ldsa;                                // lds_addr (bytes)
        g0[2] = (unsigned)(gaddr & 0xFFFFFFFFull);   // global_addr[31:0]
        g0[3] = (unsigned)((gaddr >> 32) & 0x1FFFFFFu) | (2u << 30); // type=2

        v8i g1;
        g1[0] = (int)((2u << 16)                     // data_size = 4 bytes
                    | (1u << 20)                     // pad_enable
                    | (7u << 22)                     // pad_interval: 256 dwords
                    | (1u << 25));                   // pad_amount: 2 dwords
        g1[1] = (int)((512u & 0xFFFFu) << 16);       // tensor_dim0 = 512 (lo16)
        g1[2] = (int)((512u >> 16) | (16u << 16));   // dim0 hi | tensor_dim1 = 16
        g1[3] = (int)((16u >> 16) | (512u << 16));   // dim1 hi | tile_dim0 = 512
        g1[4] = (int)(16u);                          // tile_dim1 = 16, tile_dim2 = 0
        g1[5] = (int)(512u);                         // tensor_dim0_stride = 512
        g1[6] = 0;
        g1[7] = 0;

        v4i gz = {0, 0, 0, 0};
#if __clang_major__ >= 23
        v8i gz8 = {0, 0, 0, 0, 0, 0, 0, 0};
        __builtin_amdgcn_tensor_load_to_lds(g0, g1, gz, gz, gz8, 0);
#else
        __builtin_amdgcn_tensor_load_to_lds(g0, g1, gz, gz, 0);
#endif
        __builtin_amdgcn_s_wait_tensorcnt(0);
    }
#else
    // --- Fallback: manual cooperative staging (float4), same LDS layout ----
    for (int t = tid; t < 16 * 128; t += 128) {
        const int row = t >> 7;
        const int c4  = t & 127;
        const float4 v = ((const float4*)(x + (size_t)(ib + row) * GAT_DIN))[c4];
        float* dst = &xs[xs_idx(row, c4 * 4)];
        dst[0] = v.x; dst[1] = v.y; dst[2] = v.z; dst[3] = v.w;
    }
#endif
    __syncthreads();

    const int wave    = tid >> 5;
    const int lane    = tid & 31;
    const int nloc    = lane & 15;          // M index for A, N index for B
    const int khalf   = (lane >> 4) << 1;   // 0 (lanes 0-15) or 2 (lanes 16-31)
    const int colbase = wave * 16;

    const float* wrow = W + (size_t)(colbase + nloc) * GAT_DIN;   // B col n = W row

    v8f acc = {};
    #pragma unroll 4
    for (int kk = 0; kk < GAT_DIN; kk += 4) {
        const int k  = kk + khalf;
        const int kp = xs_idx(nloc, k);
        v2f a;  a.x  = xs[kp];    a.y  = xs[kp + 1];  // A[M][k], A[M][k+1]
        v2f bm; bm.x = wrow[k];   bm.y = wrow[k + 1]; // B[k][n], B[k+1][n]
        acc = __builtin_amdgcn_wmma_f32_16x16x4_f32(false, a, false, bm,
                                                    (short)0, acc, false, false);
    }

    // D layout: VGPR r -> row r + (lane>=16 ? 8 : 0), col = lane&15.
    const int rowoff = (lane >> 4) * 8;
    const int col    = colbase + nloc;
    const float bv   = bias[col];
    #pragma unroll
    for (int r = 0; r < 8; ++r) {
        hout[(size_t)(ib + r + rowoff) * GAT_DOUT + col] = acc[r] + bv;
    }
}

// ---------------------------------------------------------------------------
// Kernel 2: e[n,h,i] = sum_f h[n,h,f] * a2[f][i],  a2[f][i] = a[2f+i].
// ---------------------------------------------------------------------------
__global__ __launch_bounds__(256) void gat_e(const float* __restrict__ h,
                                             const float* __restrict__ a,
                                             float* __restrict__ e0,
                                             float* __restrict__ e1) {
    const int idx = blockIdx.x * 256 + threadIdx.x;   // n*8 + head
    if (idx >= GAT_N * GAT_H) return;
    const float* hp = h + (size_t)(idx >> 3) * GAT_DOUT + (idx & 7) * GAT_F;
    float s0 = 0.f, s1 = 0.f;
    #pragma unroll
    for (int f = 0; f < GAT_F; ++f) {
        const float hv = hp[f];
        s0 += hv * a[2 * f];
        s1 += hv * a[2 * f + 1];
    }
    e0[idx] = s0;
    e1[idx] = s1;
}

// ---------------------------------------------------------------------------
// Kernel 3: pack adj[i][j] != 0 into bitmask words (j fastest, 32 j per word).
// 64MB read once -> 2MB written (lives in L2 for the attention kernel).
// ---------------------------------------------------------------------------
__global__ __launch_bounds__(256) void gat_pack(const int* __restrict__ adj,
                                                unsigned* __restrict__ packed) {
    const int idx = blockIdx.x * 256 + threadIdx.x;   // i*128 + (j/32)
    if (idx >= GAT_N * (GAT_N / 32)) return;
    const int4* p4 = (const int4*)(adj + (size_t)idx * 32);
    unsigned w = 0u;
    #pragma unroll
    for (int q = 0; q < 8; ++q) {
        const int4 v = p4[q];
        w |= (v.x != 0 ? 1u : 0u) << (q * 4 + 0);
        w |= (v.y != 0 ? 1u : 0u) << (q * 4 + 1);
        w |= (v.z != 0 ? 1u : 0u) << (q * 4 + 2);
        w |= (v.w != 0 ? 1u : 0u) << (q * 4 + 3);
    }
    packed[idx] = w;
}

// ---------------------------------------------------------------------------
// Kernel 4: flash-style attention + aggregation.
// Grid: 256 blocks (16 rows i each).  Block: 256 threads = 8 waves, wave = head.
// Scores materialized directly in WMMA A-layout (rank-1: e0[j] + e1[i]);
// row max/sum via one shfl_xor(16); P @ h_j via 4x WMMA f32 16x16x4.
// Ballot-uniform branch skips the O/l rescale when the row max didn't grow.
// ---------------------------------------------------------------------------
__global__ __launch_bounds__(256) void gat_attn(const float* __restrict__ h,
                                                const float* __restrict__ e0,
                                                const float* __restrict__ e1,
                                                const unsigned* __restrict__ packed,
                                                float* __restrict__ out) {
    const int ib   = blockIdx.x * 16;
    const int head = threadIdx.x >> 5;     // wave id == head
    const int lane = threadIdx.x & 31;
    const int il   = lane & 15;            // local row i (A-layout M)
    const int hi   = lane >> 4;            // half: 0 or 1
    const int off  = hi * 2;               // j-slot offset inside a K=4 chunk

    const float NEGINF = -__builtin_inff();
    const float e1i = e1[(size_t)(ib + il) * GAT_H + head];
    const unsigned* prow = packed + (size_t)(ib + il) * (GAT_N / 32);

    float m = NEGINF;   // running row max (row il; lanes il and il+16 mirror)
    float l = 0.f;      // running row sum
    v8f acc = {};       // O in C-layout; cols 8..15 stay zero

    for (int jb = 0; jb < GAT_N; jb += 16) {
        // e0 for this j-block: lanes 0-15 hold e0[jb+lane]; broadcast via shfl.
        const float e0v = e0[(size_t)(jb + il) * GAT_H + head];
        // adjacency bits for row il, j in [jb, jb+16)
        const unsigned bits = (prow[jb >> 5] >> (jb & 16)) & 0xFFFFu;

        if (jb + 32 < GAT_N) {
            __builtin_prefetch(&prow[(jb + 32) >> 5], 0, 0);            // global_prefetch_b8
            __builtin_prefetch(h + (size_t)(jb + 16) * GAT_DOUT, 0, 0);
        }

        // --- masked leaky-relu scores in A-layout slots ------------------
        float s[4][2];
        float mloc = NEGINF;
        #pragma unroll
        for (int c = 0; c < 4; ++c) {
            const int j0 = 4 * c + off;
            float v0 = __shfl(e0v, j0)     + e1i;
            float v1 = __shfl(e0v, j0 + 1) + e1i;
            v0 = (v0 > 0.f) ? v0 : GAT_ALPHA * v0;
            v1 = (v1 > 0.f) ? v1 : GAT_ALPHA * v1;
            v0 = ((bits >> j0) & 1u)       ? v0 : NEGINF;
            v1 = ((bits >> (j0 + 1)) & 1u) ? v1 : NEGINF;
            s[c][0] = v0; s[c][1] = v1;
            mloc = fmaxf(mloc, fmaxf(v0, v1));
        }
        const float rowmax = fmaxf(mloc, __shfl_xor(mloc, 16));
        const float mnew   = fmaxf(m, rowmax);
        const float mg     = (mnew == NEGINF) ? 0.f : mnew;

        // --- P = exp(s - mnew), row sum ----------------------------------
        float p[4][2];
        float sumloc = 0.f;
        #pragma unroll
        for (int c = 0; c < 4; ++c) {
            const float p0 = (s[c][0] == NEGINF) ? 0.f : __expf(s[c][0] - mg);
            const float p1 = (s[c][1] == NEGINF) ? 0.f : __expf(s[c][1] - mg);
            p[c][0] = p0; p[c][1] = p1;
            sumloc += p0 + p1;
        }
        const float rowsum = sumloc + __shfl_xor(sumloc, 16);

        // --- rescale only if any row's max grew (wave-uniform branch) ----
        if (__ballot(mnew > m) != 0ull) {
            const float scale = __expf(m - mg);    // m == -inf -> 0 (safe)
            l = l * scale + rowsum;
            #pragma unroll
            for (int r = 0; r < 8; ++r) {
                acc[r] *= __shfl(scale, r + hi * 8);  // row stats at lane == row
            }
        } else {
            l += rowsum;                           // scale == 1 exactly
        }
        m = mnew;

        // --- O += P @ h_j via 4x WMMA f32 16x16x4 ------------------------
        #pragma unroll
        for (int c = 0; c < 4; ++c) {
            v2f bm;
            if (il < GAT_F) {
                const float* hb = h + (size_t)(jb + 4 * c + off) * GAT_DOUT
                                    + head * GAT_F + il;
                bm.x = hb[0];           // B[off  ][n]
                bm.y = hb[GAT_DOUT];    // B[off+1][n]
            } else {
                bm.x = 0.f; bm.y = 0.f; // pad columns 8..15
            }
            v2f av; av.x = p[c][0]; av.y = p[c][1];
            acc = __builtin_amdgcn_wmma_f32_16x16x4_f32(false, av, false, bm,
                                                        (short)0, acc, false, false);
        }
    }

    // --- epilogue: divide by row sum, store cols 0..7 of C-layout tile ---
    const float linv = 1.f / l;
    #pragma unroll
    for (int r = 0; r < 8; ++r) {
        const float li = __shfl(linv, r + hi * 8);
        if (il < GAT_F) {
            out[(size_t)(ib + r + hi * 8) * GAT_DOUT + head * GAT_F + il] = acc[r] * li;
        }
    }
}

// ---------------------------------------------------------------------------
extern "C" void kernel_launch(void* const* d_in, const int* in_sizes, int n_in,
                              void* d_out, int out_size, void* d_ws, size_t ws_size,
                              hipStream_t stream) {
    const float* x   = (const float*)d_in[0];
    const int*   adj = (const int*)d_in[1];
    const float* W   = (const float*)d_in[2];
    const float* b   = (const float*)d_in[3];
    const float* a   = (const float*)d_in[4];
    float* out = (float*)d_out;

    char* ws = (char*)d_ws;
    float*    hbuf   = (float*)ws;                                  // 1 MB
    float*    e0     = (float*)(ws + (size_t)GAT_N * GAT_DOUT * 4); // 128 KB
    float*    e1     = e0 + GAT_N * GAT_H;                          // 128 KB
    unsigned* packed = (unsigned*)(ws + (size_t)GAT_N * GAT_DOUT * 4
                                      + 2 * (size_t)GAT_N * GAT_H * 4); // 2 MB

    gat_linear<<<GAT_N / 16, 128, 0, stream>>>(x, W, b, hbuf);
    gat_e<<<(GAT_N * GAT_H) / 256, 256, 0, stream>>>(hbuf, a, e0, e1);
    gat_pack<<<(GAT_N * (GAT_N / 32)) / 256, 256, 0, stream>>>(adj, packed);
    gat_attn<<<GAT_N / 16, 256, 0, stream>>>(hbuf, e0, e1, packed, out);
}